// GCN_66194035966386
// MI455X (gfx1250) — compile-verified
//
#include <hip/hip_runtime.h>
#include <hip/hip_bf16.h>

typedef __attribute__((ext_vector_type(2))) float v2f;
typedef __attribute__((ext_vector_type(8))) float v8f;

// ---------------------------------------------------------------------------
// Degree / normalization kernels
// ---------------------------------------------------------------------------
__global__ void deg_init_kernel(float* deg, int n) {
    int i = blockIdx.x * blockDim.x + threadIdx.x;
    if (i < n) deg[i] = 1.0f;  // self-loop weight
}

__global__ void deg_accum_kernel(const int* dst, const float* w, float* deg, int e) {
    int i = blockIdx.x * blockDim.x + threadIdx.x;
    if (i < e) atomicAdd(&deg[dst[i]], w[i]);
}

__global__ void dinv_kernel(float* deg, int n) {
    int i = blockIdx.x * blockDim.x + threadIdx.x;
    if (i < n) {
        float d = deg[i];
        deg[i] = (d > 0.0f) ? rsqrtf(d) : 0.0f;
    }
}

// ---------------------------------------------------------------------------
// GEMM via V_WMMA_F32_16X16X4_F32 (fp32 in, fp32 acc — matches reference
// precision). One wave32 computes one 16x16 output tile; K advanced by 4.
// relu_in != 0 applies ReLU to the A operand on load (deferred activation
// from the previous layer's aggregation).
//
// ISA layouts (CDNA5 ISA §7.12.2):
//   A 16x4 : lanes 0-15 -> M=lane, VGPR0=K0 VGPR1=K1; lanes 16-31 -> K2,K3
//   B 4x16 : lanes 0-15 -> N=lane, VGPR0=K0 VGPR1=K1; lanes 16-31 -> K2,K3
//   C/D    : VGPR r: lanes 0-15 -> M=r, N=lane; lanes 16-31 -> M=8+r
// ---------------------------------------------------------------------------
__global__ void __launch_bounds__(256)
gemm_wmma_f32_kernel(const float* __restrict__ A, const float* __restrict__ W,
                     float* __restrict__ Out, int M, int K, int F, int relu_in) {
    int wid  = (blockIdx.x * blockDim.x + threadIdx.x) >> 5;  // wave id
    int lane = threadIdx.x & 31;
    int tiles_n = F >> 4;
    int total   = (M >> 4) * tiles_n;
    if (wid >= total) return;           // wave-uniform: EXEC stays all-1s

    int tm  = wid / tiles_n;
    int tn  = wid - tm * tiles_n;
    int l16  = lane & 15;
    int half = lane >> 4;

    int row = tm * 16 + l16;            // A row (M) for this lane
    int col = tn * 16 + l16;            // B/D column (N) for this lane

    const float* arow = A + (size_t)row * K;
    v8f c = {0.f, 0.f, 0.f, 0.f, 0.f, 0.f, 0.f, 0.f};

    for (int k = 0; k < K; k += 4) {
        int ka = k + 2 * half;          // upper half-lanes carry K+2,K+3
        v2f a;
        a.x = arow[ka];
        a.y = arow[ka + 1];
        if (relu_in) {
            a.x = fmaxf(a.x, 0.0f);
            a.y = fmaxf(a.y, 0.0f);
        }
        v2f b;
        b.x = W[(size_t)ka * F + col];
        b.y = W[(size_t)(ka + 1) * F + col];
        // (neg_a, A, neg_b, B, c_mod, C, reuse_a, reuse_b)
        c = __builtin_amdgcn_wmma_f32_16x16x4_f32(false, a, false, b,
                                                  (short)0, c, false, false);
    }

    int mbase = tm * 16 + half * 8;     // C/D layout: upper lanes hold M=8..15
#pragma unroll
    for (int r = 0; r < 8; ++r)
        Out[(size_t)(mbase + r) * F + col] = c[r];
}

// ---------------------------------------------------------------------------
// AGG[i,f] = bias[f] + dinv[i]^2 * H[i,f]   (fused bias + self-loop term)
// ---------------------------------------------------------------------------
__global__ void init_agg_kernel(const float* __restrict__ H,
                                const float* __restrict__ bias,
                                const float* __restrict__ dinv,
                                float* __restrict__ agg, int n, int F) {
    long long i = (long long)blockIdx.x * blockDim.x + threadIdx.x;
    long long tot = (long long)n * F;
    if (i >= tot) return;
    int node = (int)(i / F);
    int f    = (int)(i - (long long)node * F);
    float di = dinv[node];
    agg[i] = bias[f] + di * di * H[i];
}

// ---------------------------------------------------------------------------
// Edge scatter: one wave32 per edge; lanes stride the feature dimension.
// agg[dst, f] += dinv[src]*w*dinv[dst] * H[src, f]  via global_atomic_add_f32
// ---------------------------------------------------------------------------
__global__ void __launch_bounds__(256)
scatter_kernel(const int* __restrict__ src, const int* __restrict__ dst,
               const float* __restrict__ w, const float* __restrict__ dinv,
               const float* __restrict__ H, float* __restrict__ agg,
               int e, int F) {
    int wid  = (blockIdx.x * blockDim.x + threadIdx.x) >> 5;
    int lane = threadIdx.x & 31;
    if (wid >= e) return;
    int s = src[wid];
    int d = dst[wid];
    float nrm = dinv[s] * w[wid] * dinv[d];
    const float* h = H + (size_t)s * F;
    float* o = agg + (size_t)d * F;
    for (int f = lane; f < F; f += 32)
        atomicAdd(&o[f], nrm * h[f]);
}

// ---------------------------------------------------------------------------
// Host launcher
// ---------------------------------------------------------------------------
extern "C" void kernel_launch(void* const* d_in, const int* in_sizes, int n_in,
                              void* d_out, int out_size, void* d_ws, size_t ws_size,
                              hipStream_t stream) {
    const float* x    = (const float*)d_in[0];
    const int*   ei   = (const int*)d_in[1];
    const float* ew   = (const float*)d_in[2];
    const float* W1   = (const float*)d_in[3];
    const float* b1   = (const float*)d_in[4];
    const float* W2   = (const float*)d_in[5];
    const float* b2   = (const float*)d_in[6];
    const float* W3   = (const float*)d_in[7];
    const float* b3   = (const float*)d_in[8];
    float*       out  = (float*)d_out;

    const int hidden  = in_sizes[4];                 // 128
    const int out_dim = in_sizes[8];                 // 64
    const int in_dim  = in_sizes[3] / hidden;        // 256
    const int n       = in_sizes[0] / in_dim;        // 50000
    const int e       = in_sizes[2];                 // 800000

    const int* src = ei;          // edge_index[0]
    const int* dst = ei + e;      // edge_index[1]

    // Workspace layout (256B aligned): dinv[n] | H[n*hidden] | AGG[n*hidden]
    char* ws = (char*)d_ws;
    size_t off = 0;
    float* dinv = (float*)(ws + off);
    off += ((size_t)n * sizeof(float) + 255) & ~(size_t)255;
    float* H = (float*)(ws + off);
    off += ((size_t)n * hidden * sizeof(float) + 255) & ~(size_t)255;
    float* AGG = (float*)(ws + off);

    const int B = 256;
    auto cdiv = [](long long a, long long b) { return (int)((a + b - 1) / b); };

    // --- normalization: deg -> dinv ---
    deg_init_kernel<<<cdiv(n, B), B, 0, stream>>>(dinv, n);
    deg_accum_kernel<<<cdiv(e, B), B, 0, stream>>>(dst, ew, dinv, e);
    dinv_kernel<<<cdiv(n, B), B, 0, stream>>>(dinv, n);

    // --- layer 1: x(256) -> hidden(128) ---
    {
        int tiles = (n / 16) * (hidden / 16);
        gemm_wmma_f32_kernel<<<cdiv((long long)tiles * 32, B), B, 0, stream>>>(
            x, W1, H, n, in_dim, hidden, 0);
        init_agg_kernel<<<cdiv((long long)n * hidden, B), B, 0, stream>>>(
            H, b1, dinv, AGG, n, hidden);
        scatter_kernel<<<cdiv((long long)e * 32, B), B, 0, stream>>>(
            src, dst, ew, dinv, H, AGG, e, hidden);
        // ReLU deferred into layer-2 GEMM A-loads
    }

    // --- layer 2: hidden -> hidden ---
    {
        int tiles = (n / 16) * (hidden / 16);
        gemm_wmma_f32_kernel<<<cdiv((long long)tiles * 32, B), B, 0, stream>>>(
            AGG, W2, H, n, hidden, hidden, 1);
        init_agg_kernel<<<cdiv((long long)n * hidden, B), B, 0, stream>>>(
            H, b2, dinv, AGG, n, hidden);
        scatter_kernel<<<cdiv((long long)e * 32, B), B, 0, stream>>>(
            src, dst, ew, dinv, H, AGG, e, hidden);
        // ReLU deferred into layer-3 GEMM A-loads
    }

    // --- layer 3: hidden -> out_dim, no ReLU, write into d_out ---
    {
        int tiles = (n / 16) * (out_dim / 16);
        gemm_wmma_f32_kernel<<<cdiv((long long)tiles * 32, B), B, 0, stream>>>(
            AGG, W3, H, n, hidden, out_dim, 1);
        init_agg_kernel<<<cdiv((long long)n * out_dim, B), B, 0, stream>>>(
            H, b3, dinv, out, n, out_dim);
        scatter_kernel<<<cdiv((long long)e * 32, B), B, 0, stream>>>(
            src, dst, ew, dinv, H, out, e, out_dim);
    }
}